// YOLOV3Loss_14697378087551
// MI455X (gfx1250) — compile-verified
//
#include <hip/hip_runtime.h>
#include <cmath>

// ---------------- constants from the reference ----------------
#define NUM_CLASSES 80
#define T_BOXES     50
#define EPSV        1e-7f

typedef __attribute__((ext_vector_type(2))) float v2f;
typedef __attribute__((ext_vector_type(8))) float v8f;

// Numerically-equivalent to reference: p = clip(sigmoid(v), eps, 1-eps);
// bce = -t*log(p) - (1-t)*log(1-p)
__device__ __forceinline__ float bce_sig(float v, float t) {
  float s = 1.0f / (1.0f + __expf(-v));
  s = fminf(fmaxf(s, EPSV), 1.0f - EPSV);
  return -t * __logf(s) - (1.0f - t) * __logf(1.0f - s);
}

// Full-wave (32-lane) f32 sum. First fold done on the matrix pipe:
// A = ones(16x4), B = {v, 0} (4x16)  ->  D[m,n] = v[n] + v[n+16] for all m.
// D vgpr0 in lane l holds s[l%16]; 4 xor-shuffles finish the 16-column sum.
// EXEC must be all ones at the call site (callers never early-exit).
__device__ __forceinline__ float wave_sum32(float v) {
#if __has_builtin(__builtin_amdgcn_wmma_f32_16x16x4_f32)
  v2f a; a.x = 1.0f; a.y = 1.0f;      // A[m,k] = 1 for all m,k
  v2f b; b.x = v;    b.y = 0.0f;      // B[0..1,n] = v per lane-half, B[2..3,n]=0
  v8f c = {};
  v8f d = __builtin_amdgcn_wmma_f32_16x16x4_f32(
      /*neg_a=*/false, a, /*neg_b=*/false, b,
      /*c_mod=*/(short)0, c, /*reuse_a=*/false, /*reuse_b=*/false);
  float s = d[0];                     // s = v[l%16] + v[l%16 + 16]
  s += __shfl_xor(s, 1);
  s += __shfl_xor(s, 2);
  s += __shfl_xor(s, 4);
  s += __shfl_xor(s, 8);
  return s;
#else
  float s = v;
  for (int m = 16; m; m >>= 1) s += __shfl_xor(s, m);
  return s;
#endif
}

// ---------------- kernel 1: clear the dense cell->record map ----------------
__global__ void init_idx(int* __restrict__ idx, int ncell) {
  int t = blockIdx.x * blockDim.x + threadIdx.x;
  int stride = gridDim.x * blockDim.x;
  for (int k = t; k < ncell; k += stride) idx[k] = -1;
}

// ---------------- kernel 2: anchor matching + sparse target build ----------
// One thread per batch image, serial over its T boxes => deterministic
// last-write-wins on cell collisions (cells never cross batches).
__global__ void build_targets(const float* __restrict__ txywh,
                              const int* __restrict__ tcls,
                              const int* __restrict__ ip,
                              int* __restrict__ idx,
                              float* __restrict__ recs,
                              int B, int H, int W) {
  int b = blockIdx.x * blockDim.x + threadIdx.x;
  if (b >= B) return;
  const float anchors[9][2] = {{10,13},{16,30},{33,23},{30,61},{62,45},
                               {59,119},{116,90},{156,198},{373,326}};
  float sx = 416.0f / (float)W, sy = 416.0f / (float)H;
  float saw[9], sah[9];
  #pragma unroll
  for (int j = 0; j < 9; ++j) { saw[j] = anchors[j][0] / sx; sah[j] = anchors[j][1] / sy; }
  int i = *ip;
  int base = 3 * (2 - i);            // ANCHOR_MASK rows: i=0->6.., 1->3.., 2->0..
  for (int t = 0; t < T_BOXES; ++t) {
    int o = (b * T_BOXES + t) * 4;
    float bx = txywh[o + 0] * W, by = txywh[o + 1] * H;
    float bw = txywh[o + 2] * W, bh = txywh[o + 3] * H;
    float area = bw * bh;
    float best = -1.0f; int aid = 0;
    #pragma unroll
    for (int j = 0; j < 9; ++j) {    // zero-centered IoU; strict '>' = first-max (argmax)
      float inter = fminf(bw, saw[j]) * fminf(bh, sah[j]);
      float iou = inter / (area + saw[j] * sah[j] - inter);
      if (iou > best) { best = iou; aid = j; }
    }
    int pos = aid - base;
    if (pos < 0 || pos > 2) continue;
    int gx = (int)floorf(bx); int gy = (int)floorf(by);
    gx = min(max(gx, 0), W - 1); gy = min(max(gy, 0), H - 1);
    int rid = b * T_BOXES + t;
    float* r = recs + rid * 8;
    r[0] = bx - (float)gx;
    r[1] = by - (float)gy;
    r[2] = __logf(bw / saw[aid]);
    r[3] = __logf(bh / sah[aid]);
    r[4] = area / (float)(H * W);    // loss_scale
    r[5] = (float)tcls[b * T_BOXES + t];
    r[6] = 0.0f; r[7] = 0.0f;
    idx[((b * 3 + pos) * H + gy) * W + gx] = rid;
  }
}

// ---------------- kernel 3: main pass (conf dense, positives sparse) -------
// Thread handles 4 consecutive x cells of one (b, anchor, y) row: one float4
// objectness load + one int4 idx load. Positive cells (<=1600 total) pull the
// remaining 84 channels with strided scalar loads.
// acc layout: 0:lx 1:ly 2:lw 3:lh 4:conf 5:cls 6:pos_count
__global__ void __launch_bounds__(256) loss_main(
    const float* __restrict__ pred, const int* __restrict__ idx,
    const float* __restrict__ recs, float* __restrict__ partials,
    int B, int H, int W) {
  const int HW = H * W;
  const int Wq = W >> 2;
  const int N4 = B * 3 * H * Wq;
  float acc[7] = {0.f, 0.f, 0.f, 0.f, 0.f, 0.f, 0.f};
  const int stride = gridDim.x * blockDim.x;
  for (int q = blockIdx.x * blockDim.x + threadIdx.x; q < N4; q += stride) {
    int x4 = (q % Wq) << 2;
    int r  = q / Wq;
    int y  = r % H;  r /= H;
    int a  = r % 3;
    int b  = r / 3;
    int cell0 = ((b * 3 + a) * H + y) * W + x4;
    int chan0 = b * 255 + a * 85;
    const float4 p4  = *(const float4*)(pred + (size_t)(chan0 + 4) * HW + y * W + x4);
    const int4   id4 = *(const int4*)(idx + cell0);
    const float pv[4] = {p4.x, p4.y, p4.z, p4.w};
    const int   iv[4] = {id4.x, id4.y, id4.z, id4.w};
    #pragma unroll
    for (int j = 0; j < 4; ++j) {
      float m = (iv[j] >= 0) ? 1.0f : 0.0f;
      acc[4] += bce_sig(pv[j], m);           // conf BCE over every cell
      acc[6] += m;                           // pos_num
      if (iv[j] >= 0) {                      // rare: full 85-channel cell
        const float* rec = recs + (size_t)iv[j] * 8;
        float tx = rec[0], ty = rec[1], tw = rec[2], th = rec[3];
        float sc = 2.0f - rec[4];
        int cls  = (int)rec[5];
        const float* pb = pred + (size_t)chan0 * HW + y * W + x4 + j;
        float v0 = pb[0];
        float v1 = pb[(size_t)HW];
        float v2 = pb[(size_t)2 * HW];
        float v3 = pb[(size_t)3 * HW];
        acc[0] += bce_sig(v0, tx) * sc;
        acc[1] += bce_sig(v1, ty) * sc;
        float dw = v2 - tw, dh = v3 - th;
        acc[2] += dw * dw * sc;
        acc[3] += dh * dh * sc;
        const float* pc = pb + (size_t)5 * HW;
        for (int c = 0; c < NUM_CLASSES; ++c) {
          float t = (c == cls) ? 1.0f : 0.0f;
          acc[5] += bce_sig(pc[(size_t)c * HW], t);
        }
      }
    }
  }
  // ---- deterministic block reduction: WMMA wave fold -> LDS across waves ----
  __shared__ float red[8][7];
  const int lane = threadIdx.x & 31;
  const int wave = threadIdx.x >> 5;
  float wsum[7];
  #pragma unroll
  for (int k = 0; k < 7; ++k) wsum[k] = wave_sum32(acc[k]);
  if (lane == 0) {
    #pragma unroll
    for (int k = 0; k < 7; ++k) red[wave][k] = wsum[k];
  }
  __syncthreads();
  if (threadIdx.x == 0) {
    const int nw = blockDim.x >> 5;
    float* p = partials + (size_t)blockIdx.x * 8;
    #pragma unroll
    for (int k = 0; k < 7; ++k) {
      float s = 0.0f;
      for (int w = 0; w < nw; ++w) s += red[w][k];
      p[k] = s;
    }
    p[7] = 0.0f;
  }
}

// ---------------- kernel 4: fold partials + final scalar math --------------
__global__ void finalize(const float* __restrict__ partials, int nb,
                         const int* __restrict__ ip, float ncell,
                         float* __restrict__ out) {
  if (blockIdx.x != 0 || threadIdx.x != 0) return;
  float s[7] = {0.f, 0.f, 0.f, 0.f, 0.f, 0.f, 0.f};
  for (int bkt = 0; bkt < nb; ++bkt) {
    const float* p = partials + (size_t)bkt * 8;
    #pragma unroll
    for (int k = 0; k < 7; ++k) s[k] += p[k];
  }
  const float balance[3] = {0.4f, 1.0f, 4.0f};
  int i = *ip;
  float pos = s[6];
  float den = fmaxf(pos, 1.0f);
  float loc = ((s[0] + s[1] + s[2] + s[3]) / den) * 0.1f;
  float cls = s[5] / (den * (float)NUM_CLASSES);
  float conf = (s[4] / ncell) * balance[i];
  bool has = pos > 0.0f;
  float total = (has ? (loc * 0.05f + cls * 0.5f) : 0.0f) + conf * 1.0f;
  out[0] = total;
  out[1] = conf;
  out[2] = has ? loc : 0.0f;
  out[3] = has ? cls : 0.0f;
}

// ---------------- launcher --------------------------------------------------
extern "C" void kernel_launch(void* const* d_in, const int* in_sizes, int n_in,
                              void* d_out, int out_size, void* d_ws, size_t ws_size,
                              hipStream_t stream) {
  const int*   ip    = (const int*)d_in[0];
  const float* pred  = (const float*)d_in[1];
  const float* txywh = (const float*)d_in[2];
  const int*   tcls  = (const int*)d_in[3];

  const int B  = in_sizes[3] / T_BOXES;             // 32
  const int HW = in_sizes[1] / (B * 255);           // 2704
  const int H  = (int)(sqrtf((float)HW) + 0.5f);    // 52
  const int W  = H;
  const int ncell = B * 3 * H * W;

  // scratch layout: [idx: ncell i32][recs: B*T*8 f32][partials: NB*8 f32]
  char* wsb = (char*)d_ws;
  int* idx = (int*)wsb;
  size_t off = ((size_t)ncell * sizeof(int) + 15) & ~(size_t)15;
  float* recs = (float*)(wsb + off);
  off += (size_t)B * T_BOXES * 8 * sizeof(float);
  float* partials = (float*)(wsb + off);

  const int NB = 240, NT = 256;
  init_idx<<<256, 256, 0, stream>>>(idx, ncell);
  build_targets<<<(B + 31) / 32, 32, 0, stream>>>(txywh, tcls, ip, idx, recs, B, H, W);
  loss_main<<<NB, NT, 0, stream>>>(pred, idx, recs, partials, B, H, W);
  finalize<<<1, 32, 0, stream>>>(partials, NB, ip, (float)ncell, (float*)d_out);
}